// GCRANEHid_58789512348193
// MI455X (gfx1250) — compile-verified
//
#include <hip/hip_runtime.h>

typedef __attribute__((ext_vector_type(2))) float v2f;
typedef __attribute__((ext_vector_type(8))) float v8f;

#define NFEAT 256
#define NDIM  128

// ---------------------------------------------------------------------------
// x1 = concat(emb_node, emb_attri) @ W1^T   via V_WMMA_F32_16X16X4_F32
// block = 256 threads = 8 wave32s; block b -> row tile b; wave w -> col tile w
// ---------------------------------------------------------------------------
__global__ __launch_bounds__(256) void gemm_x1_wmma(
    const float* __restrict__ emb_node,
    const float* __restrict__ emb_attri,
    const float* __restrict__ W1,
    float* __restrict__ x1,
    int nnode, int ntot)
{
    const int lane  = threadIdx.x & 31;
    const int ct    = threadIdx.x >> 5;     // column tile 0..7
    const int rt    = blockIdx.x;           // row tile
    const int l16   = lane & 15;
    const int khalf = (lane >> 4) << 1;     // 0 or 2 (K sub-offset per ISA layout)

    // A fragment source row (x0 row), clamped for safety at a partial tail tile
    int m = rt * 16 + l16;
    if (m >= ntot) m = ntot - 1;
    const float* arow = (m < nnode)
        ? (emb_node  + (size_t)m * NFEAT)
        : (emb_attri + (size_t)(m - nnode) * NFEAT);

    // B fragment: B[k][n] = W1[n][k]; W1 is [NDIM, NFEAT] row-major (K-major rows)
    const float* brow = W1 + (size_t)(ct * 16 + l16) * NFEAT;

    v8f c = {};
    #pragma unroll 8
    for (int k = 0; k < NFEAT; k += 4) {
        v2f a = *(const v2f*)(arow + k + khalf);
        v2f b = *(const v2f*)(brow + k + khalf);
        // (neg_a, A, neg_b, B, c_mod, C, reuse_a, reuse_b)
        c = __builtin_amdgcn_wmma_f32_16x16x4_f32(
                false, a, false, b, (short)0, c, false, false);
    }

    // D layout: VGPR r -> M = r + 8*(lane/16), N = lane%16  (within the tile)
    const int n     = ct * 16 + l16;
    const int mbase = rt * 16 + ((lane >> 4) << 3);
    float* dst = x1 + (size_t)mbase * NDIM + n;

    if (rt * 16 + 16 <= ntot) {
        // full tile: straight-line unguarded stores (common path; always taken
        // when ntot is a multiple of 16)
        #pragma unroll
        for (int r = 0; r < 8; ++r)
            dst[(size_t)r * NDIM] = c[r];
    } else {
        #pragma unroll
        for (int r = 0; r < 8; ++r)
            if (mbase + r < ntot)
                dst[(size_t)r * NDIM] = c[r];
    }
}

// ---------------------------------------------------------------------------
// zero-fill (x2, x3 regions of d_out) -- float4 grid-stride
// ---------------------------------------------------------------------------
__global__ __launch_bounds__(256) void zero_f4(float4* __restrict__ p, size_t n4)
{
    size_t i      = (size_t)blockIdx.x * blockDim.x + threadIdx.x;
    size_t stride = (size_t)gridDim.x * blockDim.x;
    float4 z; z.x = 0.f; z.y = 0.f; z.z = 0.f; z.w = 0.f;
    for (; i < n4; i += stride) p[i] = z;
}

// ---------------------------------------------------------------------------
// SpMM scatter: one wave per edge, each lane owns 4 of the 128 features.
// out[row] += val * x1[col]  via global_atomic_add_f32 (L2-resident rows).
// Edge index is forced wave-uniform so row/col/val become scalar loads.
// ---------------------------------------------------------------------------
__global__ __launch_bounds__(256) void spmm_scatter(
    const int*   __restrict__ row,
    const int*   __restrict__ col,
    const float* __restrict__ val,
    const float* __restrict__ x1,
    float*       __restrict__ out,
    int nedges)
{
    const int lane = threadIdx.x & 31;
    int e = blockIdx.x * (blockDim.x >> 5) + (threadIdx.x >> 5);
    e = __builtin_amdgcn_readfirstlane(e);      // wave-uniform -> scalar loads
    if (e >= nedges) return;

    const int   r = row[e];
    const int   s = col[e];
    const float v = val[e];

    const float4 xv = ((const float4*)(x1 + (size_t)s * NDIM))[lane];
    float* dst = out + (size_t)r * NDIM + lane * 4;

    unsafeAtomicAdd(dst + 0, xv.x * v);
    unsafeAtomicAdd(dst + 1, xv.y * v);
    unsafeAtomicAdd(dst + 2, xv.z * v);
    unsafeAtomicAdd(dst + 3, xv.w * v);
}

// ---------------------------------------------------------------------------
extern "C" void kernel_launch(void* const* d_in, const int* in_sizes, int n_in,
                              void* d_out, int out_size, void* d_ws, size_t ws_size,
                              hipStream_t stream)
{
    const float* emb_node  = (const float*)d_in[0];
    const float* emb_attri = (const float*)d_in[1];
    const float* W1        = (const float*)d_in[2];
    const int*   adj_row   = (const int*)  d_in[3];
    const int*   adj_col   = (const int*)  d_in[4];
    const float* adj_val   = (const float*)d_in[5];
    const int*   adj2_row  = (const int*)  d_in[6];
    const int*   adj2_col  = (const int*)  d_in[7];
    const float* adj2_val  = (const float*)d_in[8];

    const int nnode  = in_sizes[0] / NFEAT;
    const int nattri = in_sizes[1] / NFEAT;
    const int ntot   = nnode + nattri;          // 102000
    const int E1     = in_sizes[3];
    const int E2     = in_sizes[6];

    float* x1 = (float*)d_out;                      // [ntot, NDIM]
    float* x2 = x1 + (size_t)ntot * NDIM;
    float* x3 = x2 + (size_t)ntot * NDIM;

    // 1) zero the accumulation targets (x2 and x3 are contiguous)
    const size_t n4   = (size_t)ntot * NDIM * 2 / 4;
    const int    zblk = (int)((n4 + 255) / 256 < 32768 ? (n4 + 255) / 256 : 32768);
    zero_f4<<<zblk, 256, 0, stream>>>((float4*)x2, n4);

    // 2) x1 = x0 @ W1^T  (WMMA f32 16x16x4)
    const int rtiles = (ntot + 15) / 16;            // 6375
    gemm_x1_wmma<<<rtiles, 256, 0, stream>>>(emb_node, emb_attri, W1, x1,
                                             nnode, ntot);

    // 3) the two SpMMs (one wave per edge, 8 waves per block)
    spmm_scatter<<<(E1 + 7) / 8, 256, 0, stream>>>(adj_row, adj_col, adj_val,
                                                   x1, x2, E1);
    spmm_scatter<<<(E2 + 7) / 8, 256, 0, stream>>>(adj2_row, adj2_col, adj2_val,
                                                   x1, x3, E2);
}